// RankLoss_23184233464204
// MI455X (gfx1250) — compile-verified
//
#include <hip/hip_runtime.h>

#define MARGIN 0.2f
#define DIM    256
#define NCLS   1024   // ids are in [0,1000); pad to 1024, padded classes stay zero

typedef float v2f __attribute__((ext_vector_type(2)));
typedef float v8f __attribute__((ext_vector_type(8)));

// ---------------------------------------------------------------------------
// Kernel 1: zero the class-sum buffers and the double accumulator.
// ---------------------------------------------------------------------------
__global__ void rl_zero(float* __restrict__ buf, int n, double* __restrict__ acc) {
    int i = blockIdx.x * blockDim.x + threadIdx.x;
    if (i == 0) *acc = 0.0;
    for (; i < n; i += gridDim.x * blockDim.x) buf[i] = 0.0f;
}

// ---------------------------------------------------------------------------
// Kernel 2: per-row L2 normalize (denom = max(||x||, 1e-6)) and atomically
// scatter the normalized row into its class-sum vector. One block per row,
// 256 threads = one float per thread.
// ---------------------------------------------------------------------------
__global__ void rl_bucket(const float* __restrict__ feats,
                          const int*   __restrict__ ids,
                          float*       __restrict__ classBuf) {
    const int row = blockIdx.x;
    const int t   = threadIdx.x;   // 0..255

    float v = feats[(size_t)row * DIM + t];

    __shared__ float red[DIM];
    red[t] = v * v;
    __syncthreads();
    #pragma unroll
    for (int s = DIM / 2; s > 0; s >>= 1) {
        if (t < s) red[t] += red[t + s];
        __syncthreads();
    }
    const float denom = fmaxf(sqrtf(red[0]), 1e-6f);
    const float vn    = v / denom;

    const int c = ids[row];        // 0..999
    atomicAdd(&classBuf[(size_t)c * DIM + t], vn);   // global_atomic_add_f32
}

// ---------------------------------------------------------------------------
// Kernel 3: signed GEMM-reduce.  M = CQ (1024x256) x CD^T (256x1024), weight
// each element by (+1 on the class diagonal, -1 off it) and reduce to a
// scalar.  One wave per 16x16 tile, K-loop of 64 V_WMMA_F32_16X16X4_F32.
//
// Fragment layouts (CDNA5 ISA 7.12.2, 32-bit 16x4 A-matrix):
//   A: lane L<16  -> M=L,    x=K(k0+0), y=K(k0+1)
//      lane L>=16 -> M=L-16, x=K(k0+2), y=K(k0+3)
//   B (4x16 = CD^T tile): same striping with N=L%16 -> one float2 load from
//      the row-major CD buffer per lane per step.
//   C/D: VGPR v -> M = v + 8*(lane>=16), N = lane%16.
// ---------------------------------------------------------------------------
__global__ void rl_wmma_reduce(const float* __restrict__ CQ,
                               const float* __restrict__ CD,
                               double*      __restrict__ acc) {
    const int wave = threadIdx.x >> 5;
    const int lane = threadIdx.x & 31;
    const int tile = blockIdx.x * (blockDim.x >> 5) + wave;   // 0..4095
    const int tr   = tile >> 6;          // 64 row-tiles
    const int tc   = tile & 63;          // 64 col-tiles
    const int rowBase = tr * 16;
    const int colBase = tc * 16;

    const int lh   = lane & 15;          // M (for A) / N (for B)
    const int koff = (lane >> 4) << 1;   // 0 for lanes 0-15, 2 for lanes 16-31

    const v2f* __restrict__ Ap = (const v2f*)(CQ + (size_t)(rowBase + lh) * DIM);
    const v2f* __restrict__ Bp = (const v2f*)(CD + (size_t)(colBase + lh) * DIM);

    v8f c = {};
    #pragma unroll 8
    for (int k0 = 0; k0 < DIM; k0 += 4) {
        v2f a = Ap[(k0 + koff) >> 1];
        v2f b = Bp[(k0 + koff) >> 1];
        // 8 args: (neg_a, A, neg_b, B, c_mod, C, reuse_a, reuse_b)
        c = __builtin_amdgcn_wmma_f32_16x16x4_f32(
                false, a, false, b, (short)0, c, false, false);
    }

    // Apply +/-1 weights: +1 where global row class == col class.
    const int col   = colBase + lh;
    const int rhalf = (lane >> 4) << 3;  // 0 or 8
    float p = 0.0f;
    #pragma unroll
    for (int v = 0; v < 8; ++v) {
        const int row = rowBase + v + rhalf;
        const float e = c[v];
        p += (row == col) ? e : -e;
    }

    // wave32 reduction
    #pragma unroll
    for (int m = 16; m >= 1; m >>= 1) p += __shfl_xor(p, m, 32);

    if (lane == 0) atomicAdd(acc, (double)p);   // global_atomic_add_f64
}

// ---------------------------------------------------------------------------
// Kernel 4: finalize -> d_out[0] = MARGIN + (2*sum_match - sum_all)
// (already encoded by the +/-1 weights above).
// ---------------------------------------------------------------------------
__global__ void rl_finalize(const double* __restrict__ acc, float* __restrict__ out) {
    out[0] = MARGIN + (float)(*acc);
}

// ---------------------------------------------------------------------------
extern "C" void kernel_launch(void* const* d_in, const int* in_sizes, int n_in,
                              void* d_out, int out_size, void* d_ws, size_t ws_size,
                              hipStream_t stream) {
    (void)n_in; (void)out_size; (void)ws_size;

    const float* qf  = (const float*)d_in[0];   // [4096,256]
    const float* df  = (const float*)d_in[1];   // [8192,256]
    const int*   qid = (const int*)d_in[2];     // [4096]
    const int*   did = (const int*)d_in[3];     // [8192]
    float*       out = (float*)d_out;

    const int nq = in_sizes[0] / DIM;           // 4096
    const int nd = in_sizes[1] / DIM;           // 8192

    // Workspace layout: [0..7] double accumulator, [16..] CQ then CD.
    char*   ws     = (char*)d_ws;
    double* acc    = (double*)ws;
    float*  classQ = (float*)(ws + 16);
    float*  classD = classQ + (size_t)NCLS * DIM;

    rl_zero<<<512, 256, 0, stream>>>(classQ, 2 * NCLS * DIM, acc);
    rl_bucket<<<nq, DIM, 0, stream>>>(qf, qid, classQ);
    rl_bucket<<<nd, DIM, 0, stream>>>(df, did, classD);

    // 64x64 tiles of 16x16 = 4096 tiles; 8 waves (256 threads) per block.
    rl_wmma_reduce<<<(NCLS / 16) * (NCLS / 16) / 8, 256, 0, stream>>>(classQ, classD, acc);
    rl_finalize<<<1, 1, 0, stream>>>(acc, out);
}